// FGN_58007828300291
// MI455X (gfx1250) — compile-verified
//
#include <hip/hip_runtime.h>
#include <hip/hip_bf16.h>

// ---------------------------------------------------------------------------
// FGN forward, restructured for MI455X (gfx1250):
//   F[b,k]   = rfft(xt[b,:]) via 128x128 Cooley-Tukey -> WMMA f32 16x16x4 GEMMs
//   xf[b,k,e] = emb[e]*F[b,k]   (never materialized)
//   fourierGC = elementwise, fused into final contraction
//   irfft + emb10-matmul folded into A[k,m], B[k,m] coefficient tables
//   fc1/fc2/fc3 = WMMA GEMMs (M=16)
// Trig everywhere uses exact power-of-two phase fractions fed to V_SIN/COS_F32
// (hardware takes revolutions: sin(2*pi*x)), avoiding libm argument reduction.
// ---------------------------------------------------------------------------

typedef float v2f __attribute__((ext_vector_type(2)));
typedef float v8f __attribute__((ext_vector_type(8)));

static __device__ __forceinline__ v8f wmma4(v2f a, v2f b, v8f c) {
  // D = A(16x4) x B(4x16) + C(16x16), all f32
  return __builtin_amdgcn_wmma_f32_16x16x4_f32(false, a, false, b, (short)0, c,
                                               false, false);
}

// fr = angle / (2*pi), i.e. revolutions. V_SIN_F32 computes sin(2*pi*fr).
static __device__ __forceinline__ void sincos_rev(float fr, float* s, float* c) {
  *s = __builtin_amdgcn_sinf(fr);
  *c = __builtin_amdgcn_cosf(fr);
}

static __device__ __forceinline__ float relu_f(float v) { return fmaxf(v, 0.0f); }

static __device__ __forceinline__ float sshrink(float v) {
  const float L = 0.01f;
  return v > L ? (v - L) : (v < -L ? (v + L) : 0.0f);
}

// --- K1: 128-point DFT trig tables (symmetric): C[r,c]=cos(2*pi*r*c/128) ----
__global__ void k_trig_tables(float* __restrict__ C128, float* __restrict__ S128) {
  int i = blockIdx.x * blockDim.x + threadIdx.x;
  if (i >= 128 * 128) return;
  int r = i >> 7, c = i & 127;
  float fr = (float)((r * c) & 127) * (1.0f / 128.0f);
  float s, co;
  sincos_rev(fr, &s, &co);
  C128[i] = co;
  S128[i] = s;
}

// --- K2: irfft(+emb10 matmul) folded coefficients -------------------------
// A[k,m] =  w_k/128 * sum_{t<512} cos(2*pi*k*t/16384) * emb10[t,m]
// B[k,m] = -w_k/128 * sum_{t<512} sin(2*pi*k*t/16384) * emb10[t,m]
// w_k = 1 for k in {0, 8192}, else 2.  (sin terms vanish naturally at edges)
__global__ void k_irfft_coef(const float* __restrict__ emb10,
                             float* __restrict__ Ac, float* __restrict__ Bc) {
  int k = blockIdx.x * blockDim.x + threadIdx.x;
  if (k > 8192) return;
  float as[8], bs[8];
#pragma unroll
  for (int m = 0; m < 8; ++m) { as[m] = 0.0f; bs[m] = 0.0f; }
  for (int t = 0; t < 512; ++t) {
    int ph = (k * t) & 16383;  // exact integer phase reduction
    float s, c;
    sincos_rev((float)ph * (1.0f / 16384.0f), &s, &c);
    const float* er = emb10 + t * 8;
#pragma unroll
    for (int m = 0; m < 8; ++m) {
      as[m] = fmaf(c, er[m], as[m]);
      bs[m] = fmaf(s, er[m], bs[m]);
    }
  }
  float wk = (k == 0 || k == 8192) ? (1.0f / 128.0f) : (2.0f / 128.0f);
#pragma unroll
  for (int m = 0; m < 8; ++m) {
    Ac[k * 8 + m] = wk * as[m];
    Bc[k * 8 + m] = -wk * bs[m];
  }
}

// --- K3: FFT stage 1 (WMMA). G[k1,t2] = sum_t1 x[t1*128+t2] * W128^{t1*k1},
// then twiddle by W16384^{t2*k1}. One wave per 16x16 tile. ------------------
__global__ void k_fft_stage1(const float* __restrict__ x,
                             const float* __restrict__ C128,
                             const float* __restrict__ S128,
                             float* __restrict__ Gpr, float* __restrict__ Gpi) {
  int lane = threadIdx.x;
  int laneHi = lane >> 4, ln = lane & 15;
  int tM = blockIdx.x, tN = blockIdx.y, b = blockIdx.z;
  v8f accR = {0.f, 0.f, 0.f, 0.f, 0.f, 0.f, 0.f, 0.f};
  v8f accI = {0.f, 0.f, 0.f, 0.f, 0.f, 0.f, 0.f, 0.f};
  int am = tM * 16 + ln;    // A row (k1)
  int bt2 = tN * 16 + ln;   // B col (t2)
  const float* xb = x + (size_t)b * 16384;  // x is [B,L,N]=[16,512,32]
  for (int k0 = 0; k0 < 128; k0 += 4) {
    int ak = k0 + 2 * laneHi;
    v2f aC, aSn, bx;
    aC[0] = C128[am * 128 + ak];
    aC[1] = C128[am * 128 + ak + 1];
    aSn[0] = -S128[am * 128 + ak];   // Gi = -sum S*x
    aSn[1] = -S128[am * 128 + ak + 1];
#pragma unroll
    for (int j = 0; j < 2; ++j) {
      int t1 = ak + j;
      int t = t1 * 128 + bt2;        // time index 0..16383
      int n = t >> 9, l = t & 511;   // xt[b,t] = x[b,l,n]
      bx[j] = xb[l * 32 + n];
    }
    accR = wmma4(aC, bx, accR);
    accI = wmma4(aSn, bx, accI);
  }
  // twiddle (Gr+iGi)*(c - i s), c=cos(2*pi*t2*k1/16384)
#pragma unroll
  for (int v = 0; v < 8; ++v) {
    int k1 = tM * 16 + v + 8 * laneHi;
    int t2 = bt2;
    float s, c;
    sincos_rev((float)(k1 * t2) * (1.0f / 16384.0f), &s, &c);
    float gr = accR[v], gi = accI[v];
    size_t o = ((size_t)b * 128 + k1) * 128 + t2;
    Gpr[o] = gr * c + gi * s;
    Gpi[o] = gi * c - gr * s;
  }
}

// --- K4: FFT stage 2 (WMMA). F[k2*128+k1] = sum_t2 G'[k1,t2]*W128^{t2*k2} --
__global__ void k_fft_stage2(const float* __restrict__ Gpr,
                             const float* __restrict__ Gpi,
                             const float* __restrict__ C128,
                             const float* __restrict__ S128,
                             float* __restrict__ Fr, float* __restrict__ Fi) {
  int lane = threadIdx.x;
  int laneHi = lane >> 4, ln = lane & 15;
  int tM = blockIdx.x, tN = blockIdx.y, b = blockIdx.z;
  v8f accR = {0.f, 0.f, 0.f, 0.f, 0.f, 0.f, 0.f, 0.f};
  v8f accI = {0.f, 0.f, 0.f, 0.f, 0.f, 0.f, 0.f, 0.f};
  int am = tM * 16 + ln;   // k1
  int bn = tN * 16 + ln;   // k2
  const float* gr = Gpr + (size_t)b * 128 * 128;
  const float* gi = Gpi + (size_t)b * 128 * 128;
  for (int k0 = 0; k0 < 128; k0 += 4) {
    int ak = k0 + 2 * laneHi;  // t2 (contraction)
    v2f aR, aI, bC, bS, bSn;
#pragma unroll
    for (int j = 0; j < 2; ++j) {
      aR[j] = gr[am * 128 + ak + j];
      aI[j] = gi[am * 128 + ak + j];
      bC[j] = C128[(ak + j) * 128 + bn];  // symmetric DFT matrix
      bS[j] = S128[(ak + j) * 128 + bn];
      bSn[j] = -bS[j];
    }
    // Fr = Gr*C + Gi*S ; Fi = Gi*C - Gr*S
    accR = wmma4(aR, bC, accR);
    accR = wmma4(aI, bS, accR);
    accI = wmma4(aI, bC, accI);
    accI = wmma4(aR, bSn, accI);
  }
#pragma unroll
  for (int v = 0; v < 8; ++v) {
    int k1 = tM * 16 + v + 8 * laneHi;
    int k2 = bn;
    int k = k2 * 128 + k1;
    Fr[(size_t)b * 16384 + k] = accR[v] * 0.0078125f;  // ortho 1/sqrt(16384)
    Fi[(size_t)b * 16384 + k] = accI[v] * 0.0078125f;
  }
}

// --- K5: fused fourierGC (elementwise) + irfft/emb10 contraction ----------
// h[b,e,m] = sum_{k<=8192} Xr*A[k,m] + Xi*B[k,m], X = z(GC(emb[e]*F)) + emb[e]*F
__global__ void k_gc_contract(const float* __restrict__ Fr,
                              const float* __restrict__ Fi,
                              const float* __restrict__ emb,
                              const float* __restrict__ w0, const float* __restrict__ b0,
                              const float* __restrict__ w1, const float* __restrict__ b1,
                              const float* __restrict__ w2, const float* __restrict__ b2,
                              const float* __restrict__ Ac, const float* __restrict__ Bc,
                              float* __restrict__ h) {
  int blk = blockIdx.x;
  int b = blk >> 7, e = blk & 127;
  int tid = threadIdx.x;  // 256
  float embe = emb[e];
  float d00 = w0[e * 128 + e], d01 = w0[16384 + e * 128 + e];
  float b00 = b0[e], b01 = b0[128 + e];
  float d10 = w1[e * 128 + e], d11 = w1[16384 + e * 128 + e];
  float b10 = b1[e], b11 = b1[128 + e];
  float d20 = w2[e * 128 + e], d21 = w2[16384 + e * 128 + e];
  float b20 = b2[e], b21 = b2[128 + e];
  float acc[8];
#pragma unroll
  for (int m = 0; m < 8; ++m) acc[m] = 0.0f;
  const float* Frb = Fr + (size_t)b * 16384;
  const float* Fib = Fi + (size_t)b * 16384;
  for (int k = tid; k < 8193; k += 256) {
    float xr = embe * Frb[k], xi = embe * Fib[k];
    // layer 0
    float o_r = relu_f(xr * d00 - xi * d01 + b00);
    float o_i = relu_f(xi * d00 + xr * d01 + b01);
    float pr = sshrink(o_r), pi = sshrink(o_i);
    // layer 1 (o_i uses UPDATED o_r, matching torch op order)
    float orn = relu_f(o_r * d10 - o_i * d11 + b10);
    o_i = relu_f(o_i * d10 + orn * d11 + b11);
    o_r = orn;
    pr += sshrink(o_r);
    pi += sshrink(o_i);
    // layer 2
    orn = relu_f(o_r * d20 - o_i * d21 + b20);
    o_i = relu_f(o_i * d20 + orn * d21 + b21);
    float zr = sshrink(orn) + pr;
    float zi = sshrink(o_i) + pi;
    float Xr = zr + xr, Xi = zi + xi;  // + bias (original xf)
    const float* ac = Ac + k * 8;
    const float* bc = Bc + k * 8;
#pragma unroll
    for (int m = 0; m < 8; ++m)
      acc[m] = fmaf(Xr, ac[m], fmaf(Xi, bc[m], acc[m]));
  }
  __shared__ float red[8 * 256];
#pragma unroll
  for (int m = 0; m < 8; ++m) red[m * 256 + tid] = acc[m];
  __syncthreads();
  for (int str = 128; str > 0; str >>= 1) {
    if (tid < str) {
#pragma unroll
      for (int m = 0; m < 8; ++m) red[m * 256 + tid] += red[m * 256 + tid + str];
    }
    __syncthreads();
  }
  if (tid < 8) h[((size_t)b * 128 + e) * 8 + tid] = red[tid * 256];
}

// --- K6: fc layer (M=16) with WMMA. W is [N][K] row-major (torch Linear). --
__global__ void k_fc(const float* __restrict__ A, const float* __restrict__ W,
                     const float* __restrict__ bias, float* __restrict__ out,
                     int K, int N, int act) {
  int lane = threadIdx.x;
  int laneHi = lane >> 4, ln = lane & 15;
  int tN = blockIdx.x;
  int n = tN * 16 + ln;
  v8f acc = {0.f, 0.f, 0.f, 0.f, 0.f, 0.f, 0.f, 0.f};
  for (int k0 = 0; k0 < K; k0 += 4) {
    int ak = k0 + 2 * laneHi;
    v2f a, bm;
    a[0] = A[ln * K + ak];
    a[1] = A[ln * K + ak + 1];
    bm[0] = W[n * K + ak];   // B[k][n] = W^T
    bm[1] = W[n * K + ak + 1];
    acc = wmma4(a, bm, acc);
  }
  float bb = bias[n];
#pragma unroll
  for (int v = 0; v < 8; ++v) {
    int m = v + 8 * laneHi;
    float val = acc[v] + bb;
    if (act) val = val > 0.0f ? val : 0.01f * val;  // leaky relu 0.01
    out[m * N + n] = val;
  }
}

extern "C" void kernel_launch(void* const* d_in, const int* in_sizes, int n_in,
                              void* d_out, int out_size, void* d_ws, size_t ws_size,
                              hipStream_t stream) {
  (void)in_sizes; (void)n_in; (void)out_size; (void)ws_size;
  const float* x     = (const float*)d_in[0];   // [16,512,32]
  const float* emb   = (const float*)d_in[1];   // [1,128]
  const float* w0    = (const float*)d_in[2];   // [2,128,128]
  const float* b0    = (const float*)d_in[3];   // [2,128]
  const float* w1    = (const float*)d_in[4];
  const float* b1    = (const float*)d_in[5];
  const float* w2    = (const float*)d_in[6];
  const float* b2    = (const float*)d_in[7];
  const float* emb10 = (const float*)d_in[8];   // [512,8]
  const float* fc1w  = (const float*)d_in[9];   // [64,1024]
  const float* fc1b  = (const float*)d_in[10];  // [64]
  const float* fc2w  = (const float*)d_in[11];  // [256,64]
  const float* fc2b  = (const float*)d_in[12];  // [256]
  const float* fc3w  = (const float*)d_in[13];  // [96,256]
  const float* fc3b  = (const float*)d_in[14];  // [96]
  float* out = (float*)d_out;                   // [16,96]

  float* ws = (float*)d_ws;
  float* C128 = ws;                   // 16384
  float* S128 = C128 + 16384;         // 16384
  float* Ac   = S128 + 16384;         // 8193*8 -> 65544
  float* Bc   = Ac + 65544;           // 65544
  float* Gpr  = Bc + 65544;           // 16*128*128 = 262144
  float* Gpi  = Gpr + 262144;         // 262144
  float* Fr   = Gpi + 262144;         // 16*16384 = 262144
  float* Fi   = Fr + 262144;          // 262144
  float* h    = Fi + 262144;          // 16*1024
  float* h1   = h + 16384;            // 16*64
  float* h2   = h1 + 1024;            // 16*256

  k_trig_tables<<<64, 256, 0, stream>>>(C128, S128);
  k_irfft_coef<<<(8193 + 63) / 64, 64, 0, stream>>>(emb10, Ac, Bc);
  k_fft_stage1<<<dim3(8, 8, 16), 32, 0, stream>>>(x, C128, S128, Gpr, Gpi);
  k_fft_stage2<<<dim3(8, 8, 16), 32, 0, stream>>>(Gpr, Gpi, C128, S128, Fr, Fi);
  k_gc_contract<<<2048, 256, 0, stream>>>(Fr, Fi, emb, w0, b0, w1, b1, w2, b2,
                                          Ac, Bc, h);
  k_fc<<<4, 32, 0, stream>>>(h, fc1w, fc1b, h1, 1024, 64, 1);
  k_fc<<<16, 32, 0, stream>>>(h1, fc2w, fc2b, h2, 64, 256, 1);
  k_fc<<<6, 32, 0, stream>>>(h2, fc3w, fc3b, out, 256, 96, 0);
}